// DynamicRNN_24077586662135
// MI455X (gfx1250) — compile-verified
//
#include <hip/hip_runtime.h>
#include <hip/hip_bf16.h>

typedef float v2f __attribute__((ext_vector_type(2)));
typedef float v8f __attribute__((ext_vector_type(8)));

#define BSZ  16
#define TLEN 1024
#define DDIM 128
#define HDIM 128

// ---------------------------------------------------------------------------
// Kernel A: Z[(b*T+t), h] = sum_d X[(b*T+t), d] * Wx[d, h] + bias[h]
// M=16384, N=128, K=128 fp32 GEMM, V_WMMA_F32_16X16X4_F32.
// 64(M) x 16(N) per wave; software-pipelined fragment loads.
// ---------------------------------------------------------------------------
__global__ __launch_bounds__(128) void rnn_precompute_xwx(
    const float* __restrict__ X, const float* __restrict__ Wx,
    const float* __restrict__ bias, float* __restrict__ Z)
{
  const int wave  = blockIdx.x * (blockDim.x >> 5) + (threadIdx.x >> 5);
  const int lane  = threadIdx.x & 31;
  const int ntile = HDIM / 16;              // 8
  const int mblk  = wave / ntile;           // 0..255
  const int nt    = wave % ntile;
  const int m0    = mblk * 64;
  const int n0    = nt * 16;

  const int mrow  = lane & 15;              // A: M = lane%16
  const int khalf = (lane >> 4) * 2;        // A/B: K pair {0,1} or {2,3}
  const int ncol  = lane & 15;              // B/C/D: N = lane%16

  const float* Xr0 = X + (size_t)(m0 +  0 + mrow) * DDIM + khalf;
  const float* Xr1 = X + (size_t)(m0 + 16 + mrow) * DDIM + khalf;
  const float* Xr2 = X + (size_t)(m0 + 32 + mrow) * DDIM + khalf;
  const float* Xr3 = X + (size_t)(m0 + 48 + mrow) * DDIM + khalf;
  const float* Wp  = Wx + n0 + ncol;

  v8f acc0 = {}, acc1 = {}, acc2 = {}, acc3 = {};

  // prologue loads (k = 0)
  v2f bf; bf.x = Wp[khalf * HDIM]; bf.y = Wp[(khalf + 1) * HDIM];
  v2f a0 = *(const v2f*)(Xr0);
  v2f a1 = *(const v2f*)(Xr1);
  v2f a2 = *(const v2f*)(Xr2);
  v2f a3 = *(const v2f*)(Xr3);

  for (int k = 0; k < DDIM; k += 4) {
    const v2f cb = bf, c0 = a0, c1 = a1, c2 = a2, c3 = a3;
    if (k + 4 < DDIM) {                      // preload next iteration
      const int kb = k + 4 + khalf;
      bf.x = Wp[kb * HDIM]; bf.y = Wp[(kb + 1) * HDIM];
      a0 = *(const v2f*)(Xr0 + k + 4);
      a1 = *(const v2f*)(Xr1 + k + 4);
      a2 = *(const v2f*)(Xr2 + k + 4);
      a3 = *(const v2f*)(Xr3 + k + 4);
    }
    acc0 = __builtin_amdgcn_wmma_f32_16x16x4_f32(false, c0, false, cb, (short)0, acc0, false, false);
    acc1 = __builtin_amdgcn_wmma_f32_16x16x4_f32(false, c1, false, cb, (short)0, acc1, false, false);
    acc2 = __builtin_amdgcn_wmma_f32_16x16x4_f32(false, c2, false, cb, (short)0, acc2, false, false);
    acc3 = __builtin_amdgcn_wmma_f32_16x16x4_f32(false, c3, false, cb, (short)0, acc3, false, false);
  }

  const float bv    = bias[n0 + ncol];
  const int   rbase = m0 + ((lane >> 4) << 3);   // C/D: M = vgpr + 8*(lane/16)
  #pragma unroll
  for (int v = 0; v < 8; ++v) {
    Z[(size_t)(rbase +  0 + v) * HDIM + n0 + ncol] = acc0[v] + bv;
    Z[(size_t)(rbase + 16 + v) * HDIM + n0 + ncol] = acc1[v] + bv;
    Z[(size_t)(rbase + 32 + v) * HDIM + n0 + ncol] = acc2[v] + bv;
    Z[(size_t)(rbase + 48 + v) * HDIM + n0 + ncol] = acc3[v] + bv;
  }
}

// ---------------------------------------------------------------------------
// Kernel B: per-batch recurrence, blocked by 16 time steps.
// Block [I, I+16):  PRE = cond[b, I:I+16, 0:I] @ H[0:I, :]  (f32 WMMA,
// 8 N-tiles x 2 K-halves over 16 waves, dual accumulators, pipelined loads,
// B fragments from a K-pair-interleaved history copy -> one coalesced b64).
// Then 16 cheap sequential steps (tail agg + Wh GEMV + tanh) on VALU.
// ---------------------------------------------------------------------------
__global__ __launch_bounds__(512) void rnn_recurrent(
    const float* __restrict__ cond, const float* __restrict__ Wh,
    const float* __restrict__ Z, float* __restrict__ out, float* Hint)
{
  __shared__ float sWh[HDIM * HDIM];        // 64 KB
  __shared__ float sPREp[2][16][HDIM];      // 16 KB  (K-split WMMA partials)
  __shared__ float sPRE[16][HDIM];          //  8 KB
  __shared__ float sRecent[16][HDIM];       //  8 KB  (this block's h vectors)
  __shared__ float sCtail[16][16];          //  1 KB
  __shared__ float sAgg[HDIM];
  __shared__ float sPart[4][HDIM];

  const int b    = blockIdx.x;
  const int tid  = threadIdx.x;
  const int lane = tid & 31;
  const int wv   = tid >> 5;                // 0..15
  const int h    = tid & (HDIM - 1);
  const int qrt  = tid >> 7;                // 0..3

  // WMMA lane mapping (16x16x4 f32)
  const int mrow  = lane & 15;
  const int khalf = (lane >> 4) * 2;
  const int ncol  = lane & 15;
  const int nt    = wv & 7;                 // N-tile
  const int kh    = wv >> 3;                // K-half
  const int n0    = nt * 16;

  for (int idx = tid; idx < HDIM * HDIM; idx += 512) sWh[idx] = Wh[idx];

  const float* condb = cond + (size_t)b * TLEN * TLEN;
  const float* Zb    = Z    + (size_t)b * TLEN * HDIM;
  float*       outb  = out  + (size_t)b * TLEN * HDIM;
  float*       Hib   = Hint ? Hint + (size_t)b * TLEN * HDIM : nullptr;

  __syncthreads();

  for (int I = 0; I < TLEN; I += 16) {
    __builtin_prefetch(Zb + (size_t)I * HDIM, 0, 1);          // global_prefetch_b8

    // ---- PRE = Cblk[16 x I] @ Hist[I x 128] via f32 WMMA ----
    if (I > 0) {
      const int half = I >> 1;              // multiple of 8
      const int kbeg = kh ? half : 0;
      const int kend = kh ? I    : half;    // span is a multiple of 8
      const float* Arow = condb + (size_t)(I + mrow) * TLEN + khalf;
      v8f acc0 = {}, acc1 = {};

      if (Hib) {
        // interleaved layout: elem (k,n) at Hib[(k>>1)*256 + n*2 + (k&1)]
        const float* Bcol = Hib + (size_t)(lane >> 4) * 256 + (n0 + ncol) * 2;
        v2f a0 = *(const v2f*)(Arow + kbeg);
        v2f b0 = *(const v2f*)(Bcol + (size_t)(kbeg >> 1) * 256);
        v2f a1 = *(const v2f*)(Arow + kbeg + 4);
        v2f b1 = *(const v2f*)(Bcol + (size_t)((kbeg + 4) >> 1) * 256);
        for (int k = kbeg; k < kend; k += 8) {
          const v2f ca0 = a0, cb0 = b0, ca1 = a1, cb1 = b1;
          if (k + 8 < kend) {               // preload next pair
            a0 = *(const v2f*)(Arow + k + 8);
            b0 = *(const v2f*)(Bcol + (size_t)((k + 8)  >> 1) * 256);
            a1 = *(const v2f*)(Arow + k + 12);
            b1 = *(const v2f*)(Bcol + (size_t)((k + 12) >> 1) * 256);
          }
          acc0 = __builtin_amdgcn_wmma_f32_16x16x4_f32(false, ca0, false, cb0, (short)0, acc0, false, false);
          acc1 = __builtin_amdgcn_wmma_f32_16x16x4_f32(false, ca1, false, cb1, (short)0, acc1, false, false);
        }
      } else {
        // fallback: strided b32 loads straight from the output history
        for (int k = kbeg; k < kend; k += 8) {
          const int kb0 = k + khalf, kb1 = k + 4 + khalf;
          v2f a0 = *(const v2f*)(Arow + k);
          v2f a1 = *(const v2f*)(Arow + k + 4);
          v2f b0, b1;
          b0.x = outb[(size_t)kb0       * HDIM + n0 + ncol];
          b0.y = outb[(size_t)(kb0 + 1) * HDIM + n0 + ncol];
          b1.x = outb[(size_t)kb1       * HDIM + n0 + ncol];
          b1.y = outb[(size_t)(kb1 + 1) * HDIM + n0 + ncol];
          acc0 = __builtin_amdgcn_wmma_f32_16x16x4_f32(false, a0, false, b0, (short)0, acc0, false, false);
          acc1 = __builtin_amdgcn_wmma_f32_16x16x4_f32(false, a1, false, b1, (short)0, acc1, false, false);
        }
      }

      const v8f accs  = acc0 + acc1;
      const int rbase = (lane >> 4) << 3;
      #pragma unroll
      for (int v = 0; v < 8; ++v)
        sPREp[kh][rbase + v][n0 + ncol] = accs[v];
      __syncthreads();
      #pragma unroll
      for (int e = 0; e < 4; ++e) {
        const int idx = tid + e * 512;
        ((float*)sPRE)[idx] = ((float*)sPREp[0])[idx] + ((float*)sPREp[1])[idx];
      }
    } else {
      #pragma unroll
      for (int e = 0; e < 4; ++e) ((float*)sPRE)[tid + e * 512] = 0.f;
    }
    // triangular tail coefficients for this block
    if (tid < 256)
      sCtail[tid >> 4][tid & 15] =
          condb[(size_t)(I + (tid >> 4)) * TLEN + I + (tid & 15)];
    __syncthreads();

    // ---- 16 sequential steps ----
    for (int j = 0; j < 16; ++j) {
      const int i = I + j;
      if (tid < HDIM) {
        float ag = sPRE[j][h];
        for (int t = 0; t < j; ++t)
          ag = fmaf(sCtail[j][t], sRecent[t][h], ag);
        sAgg[h] = ag;
      }
      __syncthreads();

      // GEMV: agg @ Wh, k split across the 4 thread quarters
      float acc = 0.f;
      const int k0 = qrt * (HDIM / 4);
      #pragma unroll 8
      for (int k = 0; k < HDIM / 4; ++k)
        acc = fmaf(sAgg[k0 + k], sWh[(k0 + k) * HDIM + h], acc);
      sPart[qrt][h] = acc;
      __syncthreads();

      if (tid < HDIM) {
        const float v = tanhf(Zb[(size_t)i * HDIM + h] +
                              sPart[0][h] + sPart[1][h] + sPart[2][h] + sPart[3][h]);
        sRecent[j][h] = v;                         // for in-block tail
        outb[(size_t)i * HDIM + h] = v;            // final output
        if (Hib)                                   // K-pair-interleaved history
          Hib[(size_t)(i >> 1) * 256 + h * 2 + (i & 1)] = v;
      }
      __syncthreads();
    }
    __threadfence_block();   // publish global h/Hint writes to the WMMA readers
  }
}

// ---------------------------------------------------------------------------
extern "C" void kernel_launch(void* const* d_in, const int* in_sizes, int n_in,
                              void* d_out, int out_size, void* d_ws, size_t ws_size,
                              hipStream_t stream) {
  const float* X    = (const float*)d_in[0];   // [B,T,D]
  const float* cond = (const float*)d_in[1];   // [B,T,T]
  const float* Wx   = (const float*)d_in[2];   // [D,H]
  const float* Wh   = (const float*)d_in[3];   // [H,H]
  const float* bias = (const float*)d_in[4];   // [H]
  float* Z   = (float*)d_ws;                   // [B*T, H] = 8 MB scratch
  float* out = (float*)d_out;                  // [B,T,H]

  const size_t zElems = (size_t)BSZ * TLEN * HDIM;
  float* Hint = (ws_size >= 2 * zElems * sizeof(float)) ? (Z + zElems) : nullptr;

  // Phase 1: Z = X @ Wx + bias   (2048 waves, 4 WMMA tiles each)
  rnn_precompute_xwx<<<dim3(512), dim3(128), 0, stream>>>(X, Wx, bias, Z);
  // Phase 2: 16 independent blocked recurrences, one workgroup per batch elem
  rnn_recurrent<<<dim3(BSZ), dim3(512), 0, stream>>>(cond, Wh, Z, out, Hint);
}